// RelationPredictor_43379169690302
// MI455X (gfx1250) — compile-verified
//
#include <hip/hip_runtime.h>

// Problem constants (from reference): B=4, S=256, D=768, C=64
#define B_DIM 4
#define S_DIM 256
#define D_DIM 768
#define C_DIM 64
#define M_DIM (B_DIM * S_DIM)   // 1024 rows of the GEMM

typedef __attribute__((ext_vector_type(2))) float v2f;
typedef __attribute__((ext_vector_type(8))) float v8f;

// ---------------------------------------------------------------------------
// Phase 1: Y[m, c] = sum_d X[m, d] * W[c, d]   (M=1024, N=64, K=768)
// One wave32 computes one 16x16 tile of Y via V_WMMA_F32_16X16X4_F32.
// A-matrix (16x4 f32):  lane L -> row M = L&15, K = 2*(L>>4) + {0,1}  (2 VGPRs)
// B-matrix (4x16 f32):  lane L -> col N = L&15, K = 2*(L>>4) + {0,1}  (2 VGPRs)
// D-matrix (16x16 f32): VGPR g, lane L -> M = g + 8*(L>>4), N = L&15  (8 VGPRs)
// Both A and B fragments are contiguous float2 loads (X row-major, W row-major
// with B = W^T so B[k][n] = W[n][k]).
// ---------------------------------------------------------------------------
__global__ __launch_bounds__(128) void yproj_wmma_f32(
    const float* __restrict__ X,   // [M_DIM, D_DIM]
    const float* __restrict__ W,   // [C_DIM, D_DIM]
    float* __restrict__ Y)         // [M_DIM, C_DIM]
{
    const int wave = threadIdx.x >> 5;
    const int lane = threadIdx.x & 31;
    const int tile = blockIdx.x * 4 + wave;      // 256 tiles total
    const int rowTile = tile >> 2;               // 0..63  (M / 16)
    const int colTile = tile & 3;                // 0..3   (N / 16)
    const int row0 = rowTile * 16;
    const int n0   = colTile * 16;

    const int mn = lane & 15;                    // row for A, col for B
    const int h  = lane >> 4;                    // lane-half selects K pair

    const float* __restrict__ xrow = X + (size_t)(row0 + mn) * D_DIM + 2 * h;
    const float* __restrict__ wrow = W + (size_t)(n0  + mn) * D_DIM + 2 * h;

    v8f acc = {0.f, 0.f, 0.f, 0.f, 0.f, 0.f, 0.f, 0.f};

    #pragma unroll 8
    for (int k0 = 0; k0 < D_DIM; k0 += 4) {
        const float2 av = *(const float2*)(xrow + k0);
        const float2 bv = *(const float2*)(wrow + k0);
        v2f a; a.x = av.x; a.y = av.y;
        v2f b; b.x = bv.x; b.y = bv.y;
        // 8 args: (neg_a, A, neg_b, B, c_mod, C, reuse_a, reuse_b)
        acc = __builtin_amdgcn_wmma_f32_16x16x4_f32(
            false, a, false, b, (short)0, acc, false, false);
    }

    float* __restrict__ ybase = Y + (size_t)(row0 + 8 * h) * C_DIM + n0 + mn;
    #pragma unroll
    for (int g = 0; g < 8; ++g) {
        ybase[(size_t)g * C_DIM] = acc[g];
    }
}

// ---------------------------------------------------------------------------
// Phase 2: out[b,i,j,c] = Y[b,i,c] - Y[b,j,c] + bias[c]
// One block per (b,i). Thread t owns (j = t>>4 mod 16-strided, c = 4*(t&15)).
// Each j-iteration: block writes 16 contiguous rows x 256 B = 4 KB, fully
// coalesced 128-bit stores. Y (256 KB) stays L2/L1 resident; HBM traffic is
// essentially the 67 MB of output writes -> bandwidth-bound floor.
// ---------------------------------------------------------------------------
__global__ __launch_bounds__(256) void pairwise_expand(
    const float* __restrict__ Y,     // [M_DIM, C_DIM]
    const float* __restrict__ bias,  // [C_DIM]
    float* __restrict__ out)         // [M_DIM, S_DIM, C_DIM]
{
    const int bi = blockIdx.x;           // b*S + i, 0..1023
    const int b  = bi >> 8;              // bi / S_DIM
    const int t  = threadIdx.x;
    const int c4 = (t & 15) << 2;        // 0,4,...,60
    const int j0 = t >> 4;               // 0..15

    float4 yi = *(const float4*)(Y + (size_t)bi * C_DIM + c4);
    const float4 bb = *(const float4*)(bias + c4);
    yi.x += bb.x; yi.y += bb.y; yi.z += bb.z; yi.w += bb.w;

    const float* __restrict__ Yb = Y + (size_t)b * S_DIM * C_DIM;
    float* __restrict__ outRow   = out + (size_t)bi * S_DIM * C_DIM + c4;

    #pragma unroll 4
    for (int j = j0; j < S_DIM; j += 16) {
        const float4 yj = *(const float4*)(Yb + (size_t)j * C_DIM + c4);
        float4 r;
        r.x = yi.x - yj.x;
        r.y = yi.y - yj.y;
        r.z = yi.z - yj.z;
        r.w = yi.w - yj.w;
        *(float4*)(outRow + (size_t)j * C_DIM) = r;
    }
}

extern "C" void kernel_launch(void* const* d_in, const int* in_sizes, int n_in,
                              void* d_out, int out_size, void* d_ws, size_t ws_size,
                              hipStream_t stream) {
    (void)in_sizes; (void)n_in; (void)out_size; (void)ws_size;

    const float* x    = (const float*)d_in[0];  // [4,256,768]
    const float* W    = (const float*)d_in[1];  // [64,768]
    const float* bias = (const float*)d_in[2];  // [64]
    float*       out  = (float*)d_out;          // [4,256,256,64]
    float*       Y    = (float*)d_ws;           // [1024,64] = 256 KB scratch

    // Phase 1: 256 wave-tiles = 64 blocks x 4 waves (128 threads, wave32)
    yproj_wmma_f32<<<64, 128, 0, stream>>>(x, W, Y);

    // Phase 2: one block per (b,i); streams the 67 MB output at HBM rate
    pairwise_expand<<<M_DIM, 256, 0, stream>>>(Y, bias, out);
}